// AttentionalPropagation_77661598646839
// MI455X (gfx1250) — compile-verified
//
#include <hip/hip_runtime.h>
#include <hip/hip_bf16.h>
#include <math.h>

// AttentionalPropagation for MI455X (gfx1250), fp32 end-to-end using
// V_WMMA_F32_16X16X4_F32 for every GEMM + flash attention.
// Workspace requirement: 4*B*D*N + 1024 floats  ~= 64 MB.

typedef __attribute__((ext_vector_type(2))) float v2f;
typedef __attribute__((ext_vector_type(8))) float v8f;

#define B_  8
#define D_  256
#define N_  2048
#define H_  4
#define HD_ 64
#define D2_ 512

__device__ __forceinline__ v8f wmma4(v2f a, v2f b, v8f c) {
  // (neg_a, A, neg_b, B, c_mod, C, reuse_a, reuse_b)
  return __builtin_amdgcn_wmma_f32_16x16x4_f32(false, a, false, b, (short)0, c,
                                               false, false);
}

// ---------------------------------------------------------------------------
// Generic WMMA-tiled GEMM: C[b, 0..M, 0..N_] = W[M,K] * X(b)[K,N_] + bias
// Each wave computes a 16(M) x 64(N) tile.  blockDim = 128 (4 waves).
// IN_MODE:  0 = plain  Xa[(b*K+i)*N_+n]
//           1 = concat (i<K/2 -> Xa else Xb, each [b][K/2][N_])
//           2 = head gather from o[b][h][n][d], i -> (h=i%4, d=i/4)
//           3 = plain + fused BN affine + ReLU (scale/shift per channel i)
// OUT_MODE: 0 = plain C[(b*M+c)*N_+n]
//           1 = v-heads layout C[(((b*4)+c%4)*N_+n)*64 + c/4]
// ---------------------------------------------------------------------------
template <int IN_MODE, int OUT_MODE>
__global__ void __launch_bounds__(128)
proj_kernel(const float* __restrict__ W, const float* __restrict__ bias,
            const float* __restrict__ Xa, const float* __restrict__ Xb,
            const float* __restrict__ scale, const float* __restrict__ shift,
            float* __restrict__ C, int M, int K)
{
  const int lane = threadIdx.x & 31;
  const int wave = threadIdx.x >> 5;
  const int l16  = lane & 15;
  const int half = lane >> 4;
  const int asel = half * 2;          // K-pair select for A/B frags
  const int b    = blockIdx.z;
  const int c0   = blockIdx.y * 16;
  const int n0   = (blockIdx.x * 4 + wave) * 64;

  auto loadX = [&](int i, int n) -> float {
    if (IN_MODE == 0) {
      return Xa[((size_t)b * K + i) * N_ + n];
    } else if (IN_MODE == 1) {
      const int Kh = K >> 1;
      const float* p = (i < Kh) ? Xa : Xb;
      const int ii = (i < Kh) ? i : i - Kh;
      return p[((size_t)b * Kh + ii) * N_ + n];
    } else if (IN_MODE == 2) {
      const int h = i & (H_ - 1);
      const int d = i >> 2;
      return Xa[(((size_t)b * H_ + h) * N_ + n) * HD_ + d];
    } else {
      float v = Xa[((size_t)b * K + i) * N_ + n];
      v = fmaf(v, scale[i], shift[i]);
      return v > 0.f ? v : 0.f;
    }
  };

  v8f acc[4] = {};
  const int arow = c0 + l16;          // A: M = lane%16 for both halves

  for (int i0 = 0; i0 < K; i0 += 4) {
    v2f a;
    a.x = W[(size_t)arow * K + i0 + asel];
    a.y = W[(size_t)arow * K + i0 + asel + 1];
#pragma unroll
    for (int t = 0; t < 4; ++t) {
      const int n = n0 + t * 16 + l16;
      v2f bb;
      bb.x = loadX(i0 + asel + 0, n);
      bb.y = loadX(i0 + asel + 1, n);
      acc[t] = wmma4(a, bb, acc[t]);
    }
  }

#pragma unroll
  for (int r = 0; r < 8; ++r) {
    const int mrow = c0 + r + 8 * half;   // C: M = r + 8*(lane>=16)
    const float bi = bias[mrow];
#pragma unroll
    for (int t = 0; t < 4; ++t) {
      const int n = n0 + t * 16 + l16;
      const float v = acc[t][r] + bi;
      if (OUT_MODE == 0) {
        C[((size_t)b * M + mrow) * N_ + n] = v;
      } else {
        const int h = mrow & (H_ - 1);
        const int d = mrow >> 2;
        C[(((size_t)b * H_ + h) * N_ + n) * HD_ + d] = v;
      }
    }
  }
}

// ---------------------------------------------------------------------------
// Flash attention, one wave (32 threads) per 16 query rows of one (b,h).
// q,k: plain [b][c][n] with c = d*H + h.  vh: [b][h][m][d].  o: [b][h][n][d].
// ---------------------------------------------------------------------------
__global__ void __launch_bounds__(32)
attn_kernel(const float* __restrict__ q, const float* __restrict__ k,
            const float* __restrict__ vh, float* __restrict__ o)
{
  __shared__ float lds_p[16][17];     // C-layout -> A-layout transpose slab
  const int lane = threadIdx.x & 31;
  const int l16  = lane & 15;
  const int half = lane >> 4;
  const int n0 = blockIdx.x * 16;
  const int h  = blockIdx.y;
  const int b  = blockIdx.z;

  // Q A-fragments for all 16 K-steps of the 64-deep head dim, pre-scaled.
  v2f qa[16];
  const float scl = 0.125f;           // 1/sqrt(64)
#pragma unroll
  for (int ks = 0; ks < 16; ++ks) {
    const int d0 = ks * 4 + half * 2;
    qa[ks].x = q[((size_t)b * D_ + (d0 + 0) * H_ + h) * N_ + n0 + l16] * scl;
    qa[ks].y = q[((size_t)b * D_ + (d0 + 1) * H_ + h) * N_ + n0 + l16] * scl;
  }

  float mrow[8], lrow[8];
#pragma unroll
  for (int r = 0; r < 8; ++r) { mrow[r] = -3.0e38f; lrow[r] = 0.f; }
  v8f acc[4] = {};

  for (int m0 = 0; m0 < N_; m0 += 16) {
    // ---- S = (Q*scl) K^T : 16 chained WMMAs ----
    v8f s = {};
#pragma unroll
    for (int ks = 0; ks < 16; ++ks) {
      const int d0 = ks * 4 + half * 2;
      v2f kb;
      kb.x = k[((size_t)b * D_ + (d0 + 0) * H_ + h) * N_ + m0 + l16];
      kb.y = k[((size_t)b * D_ + (d0 + 1) * H_ + h) * N_ + m0 + l16];
      s = wmma4(qa[ks], kb, s);
    }
    // ---- online softmax update (row reductions across 16-lane halves) ----
#pragma unroll
    for (int r = 0; r < 8; ++r) {
      float v = s[r];
      v = fmaxf(v, __shfl_xor(v, 1, 32));
      v = fmaxf(v, __shfl_xor(v, 2, 32));
      v = fmaxf(v, __shfl_xor(v, 4, 32));
      v = fmaxf(v, __shfl_xor(v, 8, 32));
      const float mn = fmaxf(mrow[r], v);
      const float alpha = __expf(mrow[r] - mn);
      mrow[r] = mn;
      const float p = __expf(s[r] - mn);
      float rs = p;
      rs += __shfl_xor(rs, 1, 32);
      rs += __shfl_xor(rs, 2, 32);
      rs += __shfl_xor(rs, 4, 32);
      rs += __shfl_xor(rs, 8, 32);
      lrow[r] = lrow[r] * alpha + rs;
#pragma unroll
      for (int t = 0; t < 4; ++t) acc[t][r] *= alpha;
      lds_p[r + 8 * half][l16] = p;   // stash P in (row, col) form
    }
    __syncthreads();                  // 1-wave WG: ordering only (S_NOP)
    // ---- O += P * V : 4 K-steps x 4 d-tiles ----
#pragma unroll
    for (int kk = 0; kk < 4; ++kk) {
      const int kr = kk * 4 + half * 2;
      v2f pa;                         // A-frag of P from LDS
      pa.x = lds_p[l16][kr + 0];
      pa.y = lds_p[l16][kr + 1];
#pragma unroll
      for (int t = 0; t < 4; ++t) {
        const int d = t * 16 + l16;
        v2f vb;                       // coalesced: lane-contiguous in d
        vb.x = vh[(((size_t)b * H_ + h) * N_ + m0 + kr + 0) * HD_ + d];
        vb.y = vh[(((size_t)b * H_ + h) * N_ + m0 + kr + 1) * HD_ + d];
        acc[t] = wmma4(pa, vb, acc[t]);
      }
    }
    __syncthreads();
  }

#pragma unroll
  for (int r = 0; r < 8; ++r) {
    const float inv = 1.f / lrow[r];
    const int n = n0 + r + 8 * half;
#pragma unroll
    for (int t = 0; t < 4; ++t) {
      o[(((size_t)b * H_ + h) * N_ + n) * HD_ + t * 16 + l16] = acc[t][r] * inv;
    }
  }
}

// ---------------------------------------------------------------------------
// BatchNorm training stats per channel over (B,N); folds gamma/beta/mean/var
// into per-channel scale/shift for the fused final GEMM.
// ---------------------------------------------------------------------------
__global__ void __launch_bounds__(256)
bn_stats_kernel(const float* __restrict__ hbuf, const float* __restrict__ gamma,
                const float* __restrict__ beta, float* __restrict__ scale,
                float* __restrict__ shift)
{
  __shared__ float ssum[256];
  __shared__ float ssq[256];
  const int c = blockIdx.x;
  float s = 0.f, q = 0.f;
  for (int idx = threadIdx.x; idx < B_ * N_; idx += 256) {
    const int b = idx >> 11;
    const int n = idx & (N_ - 1);
    const float v = hbuf[((size_t)b * D2_ + c) * N_ + n];
    s += v;
    q += v * v;
  }
  ssum[threadIdx.x] = s;
  ssq[threadIdx.x]  = q;
  __syncthreads();
  for (int st = 128; st > 0; st >>= 1) {
    if ((int)threadIdx.x < st) {
      ssum[threadIdx.x] += ssum[threadIdx.x + st];
      ssq[threadIdx.x]  += ssq[threadIdx.x + st];
    }
    __syncthreads();
  }
  if (threadIdx.x == 0) {
    const float inv  = 1.f / (float)(B_ * N_);
    const float mean = ssum[0] * inv;
    const float var  = ssq[0] * inv - mean * mean;   // biased (BN training)
    const float sc   = gamma[c] * rsqrtf(var + 1e-3f);
    scale[c] = sc;
    shift[c] = beta[c] - mean * sc;
  }
}

// ---------------------------------------------------------------------------
extern "C" void kernel_launch(void* const* d_in, const int* in_sizes, int n_in,
                              void* d_out, int out_size, void* d_ws, size_t ws_size,
                              hipStream_t stream)
{
  (void)in_sizes; (void)n_in; (void)out_size; (void)ws_size;
  const float* x   = (const float*)d_in[0];
  const float* src = (const float*)d_in[1];
  const float* Wq  = (const float*)d_in[2];
  const float* bq  = (const float*)d_in[3];
  const float* Wk  = (const float*)d_in[4];
  const float* bk  = (const float*)d_in[5];
  const float* Wv  = (const float*)d_in[6];
  const float* bv  = (const float*)d_in[7];
  const float* Wm  = (const float*)d_in[8];
  const float* bm  = (const float*)d_in[9];
  const float* W1  = (const float*)d_in[10];
  const float* b1  = (const float*)d_in[11];
  const float* g1  = (const float*)d_in[12];
  const float* be1 = (const float*)d_in[13];
  const float* W2  = (const float*)d_in[14];
  const float* b2  = (const float*)d_in[15];
  float* out = (float*)d_out;

  const size_t SZ = (size_t)B_ * D_ * N_;   // 4 Mi floats
  float* buf0 = (float*)d_ws;               // q (plain)
  float* buf1 = buf0 + SZ;                  // k (plain) -> later message
  float* buf2 = buf1 + SZ;                  // v (head layout); later h[0:SZ]
  float* buf3 = buf2 + SZ;                  // attn out (head layout); later h[SZ:2SZ]
  float* hbuf = buf2;                       // h spans buf2..buf3 (B*512*N)
  float* scl  = buf3 + SZ;
  float* shf  = scl + D2_;

  const dim3 blk(128);
  // q = Wq x + bq ; k = Wk s + bk ; v = Wv s + bv (v in [b][h][m][d])
  proj_kernel<0, 0><<<dim3(8, 16, 8), blk, 0, stream>>>(Wq, bq, x,   nullptr, nullptr, nullptr, buf0, D_, D_);
  proj_kernel<0, 0><<<dim3(8, 16, 8), blk, 0, stream>>>(Wk, bk, src, nullptr, nullptr, nullptr, buf1, D_, D_);
  proj_kernel<0, 1><<<dim3(8, 16, 8), blk, 0, stream>>>(Wv, bv, src, nullptr, nullptr, nullptr, buf2, D_, D_);
  // flash attention -> o[b][h][n][d]
  attn_kernel<<<dim3(N_ / 16, H_, B_), dim3(32), 0, stream>>>(buf0, buf1, buf2, buf3);
  // message = Wm attn + bm (gather from head layout; k buffer is dead)
  proj_kernel<2, 0><<<dim3(8, 16, 8), blk, 0, stream>>>(Wm, bm, buf3, nullptr, nullptr, nullptr, buf1, D_, D_);
  // h = W1 [x; message] + b1 (v/o buffers dead -> reuse as h)
  proj_kernel<1, 0><<<dim3(8, 32, 8), blk, 0, stream>>>(W1, b1, x, buf1, nullptr, nullptr, hbuf, D2_, D2_);
  // BN training stats folded to scale/shift
  bn_stats_kernel<<<dim3(D2_), dim3(256), 0, stream>>>(hbuf, g1, be1, scl, shf);
  // out = W2 relu(BN(h)) + b2, BN+ReLU fused into the B-fragment loads
  proj_kernel<3, 0><<<dim3(8, 16, 8), blk, 0, stream>>>(W2, b2, hbuf, nullptr, scl, shf, out, D_, D2_);
}